// Quantizer_85383949845019
// MI455X (gfx1250) — compile-verified
//
#include <hip/hip_runtime.h>
#include <cstdint>

namespace {

constexpr int   kGroup         = 128;
constexpr int   kLanes         = 32;                // wave32 on gfx1250
constexpr int   kGroupsPerWave = 2;                 // lanes 0-15 -> gA, 16-31 -> gB
constexpr int   kElemsPerLane  = kGroup * kGroupsPerWave / kLanes;  // 8
constexpr int   kWavesPerBlock = 8;
constexpr int   kThreads       = kWavesPerBlock * kLanes;           // 256
constexpr int   kGroupsPerBlk  = kWavesPerBlock * kGroupsPerWave;   // 16
constexpr int   kSteps         = 80;                // int(max_shrink * grid)
constexpr float kMaxQ          = 15.0f;             // 2^4 - 1, asym

// ds_swizzle_b32 group-of-32 XOR patterns: offset = (xor<<10) | (0<<5) | 0x1f
template <int PATTERN>
__device__ __forceinline__ float swzXor(float v) {
    return __int_as_float(__builtin_amdgcn_ds_swizzle(__float_as_int(v), PATTERN));
}

// Butterfly over each 16-lane half; every lane ends with its half's result.
__device__ __forceinline__ float halfSum(float v) {
    v += swzXor<0x041F>(v);   // xor 1
    v += swzXor<0x081F>(v);   // xor 2
    v += swzXor<0x101F>(v);   // xor 4
    v += swzXor<0x201F>(v);   // xor 8
    return v;
}
__device__ __forceinline__ float halfMin(float v) {
    v = fminf(v, swzXor<0x041F>(v));
    v = fminf(v, swzXor<0x081F>(v));
    v = fminf(v, swzXor<0x101F>(v));
    v = fminf(v, swzXor<0x201F>(v));
    return v;
}
__device__ __forceinline__ float halfMax(float v) {
    v = fmaxf(v, swzXor<0x041F>(v));
    v = fmaxf(v, swzXor<0x081F>(v));
    v = fmaxf(v, swzXor<0x101F>(v));
    v = fmaxf(v, swzXor<0x201F>(v));
    return v;
}

__global__ __launch_bounds__(kThreads) void quant4_clipopt_kernel(
        const float* __restrict__ x, float* __restrict__ out, int groups) {
    __shared__ __align__(16) float tile[kWavesPerBlock * kGroupsPerWave * kGroup];

    const int lane = threadIdx.x & (kLanes - 1);
    const int wave = threadIdx.x >> 5;
    const int g0   = (blockIdx.x * kWavesPerBlock + wave) * kGroupsPerWave;
    if (g0 >= groups) return;

    int myg = g0 + (lane >> 4);                 // this half-wave's group
    if (myg >= groups) myg = groups - 1;        // benign duplicate on odd tails

    // ---- CDNA5 async copy: global -> LDS, 32 B per lane (1 KB per wave) ----
    const uint64_t gbase = (uint64_t)(uintptr_t)x;
    const uint32_t goff  =
        ((uint32_t)myg * (uint32_t)kGroup + (uint32_t)(lane & 15) * (uint32_t)kElemsPerLane) *
        (uint32_t)sizeof(float);
    float* ldsp = &tile[(wave * kLanes + lane) * kElemsPerLane];
    // Low 32 bits of the flat LDS address are the LDS byte offset (ISA 10.2).
    const uint32_t lds_off = (uint32_t)(uintptr_t)ldsp;
    // INST_OFFSET is added to both the LDS and global addresses (ISA 8.2).
    asm volatile("global_load_async_to_lds_b128 %0, %1, %2\n\t"
                 "global_load_async_to_lds_b128 %0, %1, %2 offset:16"
                 :: "v"(lds_off), "v"(goff), "s"(gbase)
                 : "memory");
    asm volatile("s_wait_asynccnt 0" ::: "memory");

    const float4 va = *reinterpret_cast<const float4*>(ldsp);
    const float4 vb = *reinterpret_cast<const float4*>(ldsp + 4);
    float xv[kElemsPerLane] = {va.x, va.y, va.z, va.w, vb.x, vb.y, vb.z, vb.w};

    // ---- per-group min/max (clamped through zero) via 16-lane XOR butterflies ----
    float lmin = xv[0], lmax = xv[0];
#pragma unroll
    for (int e = 1; e < kElemsPerLane; ++e) {
        lmin = fminf(lmin, xv[e]);
        lmax = fmaxf(lmax, xv[e]);
    }
    const float xmin = fminf(halfMin(lmin), 0.0f);
    const float xmax = fmaxf(halfMax(lmax), 0.0f);
    const bool  deg  = (xmin == 0.0f) && (xmax == 0.0f);

    // ---- 80-step clip grid search, entirely in registers ----
    float best      = __builtin_inff();
    float bestScale = 1.0f;
    float bestZero  = 0.0f;

#pragma unroll 2
    for (int i = 0; i < kSteps; ++i) {
        const float p = 1.0f - (float)i * 0.01f;
        const float a = deg ? -1.0f : p * xmin;   // _find_params asym branch
        const float b = deg ?  1.0f : p * xmax;
        const float scale = (b - a) * (1.0f / kMaxQ);
        const float invS  = __builtin_amdgcn_rcpf(scale);   // v_rcp_f32
        const float zero  = rintf(-a * invS);               // v_rndne (half-even)
        // fold "+zero ... clip(0,15) ... -zero" into shifted clamp bounds
        const float lo = -zero;
        const float hi = kMaxQ - zero;

        float acc = 0.0f;
#pragma unroll
        for (int e = 0; e < kElemsPerLane; ++e) {
            float r = rintf(xv[e] * invS);
            r = fminf(fmaxf(r, lo), hi);
            const float d = fmaf(r, -scale, xv[e]);          // x - scale*(q-zero)
            const float t = __builtin_amdgcn_logf(fabsf(d)); // log2|d| (v_log_f32)
            acc += __builtin_amdgcn_exp2f(2.4f * t);         // |d|^2.4 (v_exp_f32)
        }
        // sum over this half's 128-element group; ^(1/2.4) monotone -> compare sums
        const float err = halfSum(acc);

        const bool better = err < best;                      // strict <, inf init
        best      = better ? err   : best;
        bestScale = better ? scale : bestScale;
        bestZero  = better ? zero  : bestZero;
    }

    // ---- final quant-dequant with winning params ----
    const float invS = __builtin_amdgcn_rcpf(bestScale);
    const float lo = -bestZero;
    const float hi = kMaxQ - bestZero;
    float ov[kElemsPerLane];
#pragma unroll
    for (int e = 0; e < kElemsPerLane; ++e) {
        float r = rintf(xv[e] * invS);
        r = fminf(fmaxf(r, lo), hi);
        ov[e] = bestScale * r;
    }
    float* op = out + (size_t)myg * kGroup + (size_t)(lane & 15) * kElemsPerLane;
    *reinterpret_cast<float4*>(op)     = make_float4(ov[0], ov[1], ov[2], ov[3]);
    *reinterpret_cast<float4*>(op + 4) = make_float4(ov[4], ov[5], ov[6], ov[7]);
}

}  // namespace

extern "C" void kernel_launch(void* const* d_in, const int* in_sizes, int n_in,
                              void* d_out, int out_size, void* d_ws, size_t ws_size,
                              hipStream_t stream) {
    const float* x   = (const float*)d_in[0];
    float*       out = (float*)d_out;
    const int n      = in_sizes[0];          // 4096 * 8192
    const int groups = n / kGroup;           // 262144
    const int blocks = (groups + kGroupsPerBlk - 1) / kGroupsPerBlk;
    quant4_clipopt_kernel<<<blocks, kThreads, 0, stream>>>(x, out, groups);
}